// SEIRSDEModel_17506286698909
// MI455X (gfx1250) — compile-verified
//
#include <hip/hip_runtime.h>

// ---------------------------------------------------------------------------
// Batched SEIR Euler integration, B = 2,097,152 trajectories x 1000 steps.
// Compute-bound (~1.26e10 f32 ops vs ~59 MB traffic). Strategy: keep all state
// in VGPRs, minimize per-step ops to 6/trajectory, and execute them on the
// CDNA5 packed-f32 VALU path (v_pk_fma_f32 / v_pk_mul_f32) => 3 packed
// instructions per trajectory-step. 8 trajectories per thread for ILP.
// ---------------------------------------------------------------------------

typedef __attribute__((ext_vector_type(2))) float v2f;

#define SEIR_STEPS 1000
#define SEIR_DT 0.01f
#define SEIR_DT_OVER_N 1.0e-8f   // DT / N_POP = 0.01 / 1e6
#define NPAIR 4                  // 4 packed pairs = 8 trajectories per thread

static __device__ __forceinline__ v2f mkv2(float a, float b) {
    v2f r; r.x = a; r.y = b; return r;
}

// d = a * b (two f32 lanes per instruction)
static __device__ __forceinline__ v2f pk_mul(v2f a, v2f b) {
    v2f d;
    asm("v_pk_mul_f32 %0, %1, %2" : "=v"(d) : "v"(a), "v"(b));
    return d;
}
// d = a * b + c
static __device__ __forceinline__ v2f pk_fma(v2f a, v2f b, v2f c) {
    v2f d;
    asm("v_pk_fma_f32 %0, %1, %2, %3" : "=v"(d) : "v"(a), "v"(b), "v"(c));
    return d;
}
// d = (-a) * b + c   (neg source modifier fuses the subtraction)
static __device__ __forceinline__ v2f pk_fma_n0(v2f a, v2f b, v2f c) {
    v2f d;
    asm("v_pk_fma_f32 %0, %1, %2, %3 neg_lo:[1,0,0] neg_hi:[1,0,0]"
        : "=v"(d) : "v"(a), "v"(b), "v"(c));
    return d;
}

__global__ __launch_bounds__(256) void seir_pk_kernel(
    const float* __restrict__ beta, const float* __restrict__ sigma,
    const float* __restrict__ gamma, const float* __restrict__ S0,
    const float* __restrict__ E0,   const float* __restrict__ I0,
    float* __restrict__ out, int nthreads)
{
    const int gid = blockIdx.x * blockDim.x + threadIdx.x;
    if (gid >= nthreads) return;

    const long base4 = (long)gid * 2;   // two float4's = 8 trajectories

    // Coalesced b128 loads of all per-trajectory parameters/state.
    const float4 b_lo = ((const float4*)beta )[base4], b_hi = ((const float4*)beta )[base4 + 1];
    const float4 s_lo = ((const float4*)sigma)[base4], s_hi = ((const float4*)sigma)[base4 + 1];
    const float4 g_lo = ((const float4*)gamma)[base4], g_hi = ((const float4*)gamma)[base4 + 1];
    const float4 S_lo = ((const float4*)S0   )[base4], S_hi = ((const float4*)S0   )[base4 + 1];
    const float4 E_lo = ((const float4*)E0   )[base4], E_hi = ((const float4*)E0   )[base4 + 1];
    const float4 I_lo = ((const float4*)I0   )[base4], I_hi = ((const float4*)I0   )[base4 + 1];

    v2f S[NPAIR], E[NPAIR], I[NPAIR];
    S[0] = mkv2(S_lo.x, S_lo.y); S[1] = mkv2(S_lo.z, S_lo.w);
    S[2] = mkv2(S_hi.x, S_hi.y); S[3] = mkv2(S_hi.z, S_hi.w);
    E[0] = mkv2(E_lo.x, E_lo.y); E[1] = mkv2(E_lo.z, E_lo.w);
    E[2] = mkv2(E_hi.x, E_hi.y); E[3] = mkv2(E_hi.z, E_hi.w);
    I[0] = mkv2(I_lo.x, I_lo.y); I[1] = mkv2(I_lo.z, I_lo.w);
    I[2] = mkv2(I_hi.x, I_hi.y); I[3] = mkv2(I_hi.z, I_hi.w);

    v2f bv[NPAIR], sv[NPAIR], gv[NPAIR];
    bv[0] = mkv2(b_lo.x, b_lo.y); bv[1] = mkv2(b_lo.z, b_lo.w);
    bv[2] = mkv2(b_hi.x, b_hi.y); bv[3] = mkv2(b_hi.z, b_hi.w);
    sv[0] = mkv2(s_lo.x, s_lo.y); sv[1] = mkv2(s_lo.z, s_lo.w);
    sv[2] = mkv2(s_hi.x, s_hi.y); sv[3] = mkv2(s_hi.z, s_hi.w);
    gv[0] = mkv2(g_lo.x, g_lo.y); gv[1] = mkv2(g_lo.z, g_lo.w);
    gv[2] = mkv2(g_hi.x, g_hi.y); gv[3] = mkv2(g_hi.z, g_hi.w);

    // Per-trajectory loop-invariant coefficients (hoisted out of 1000 steps):
    //   bdtn = beta*DT/N, sdt = sigma*DT, oms = 1-sigma*DT, omg = 1-gamma*DT
    v2f bdtn[NPAIR], sdt[NPAIR], oms[NPAIR], omg[NPAIR];
#pragma unroll
    for (int k = 0; k < NPAIR; ++k) {
        bdtn[k] = bv[k] * SEIR_DT_OVER_N;
        sdt[k]  = sv[k] * SEIR_DT;
        oms[k]  = 1.0f - sdt[k];
        omg[k]  = 1.0f - gv[k] * SEIR_DT;
    }

    // Euler loop. Per pair per step (2 trajectories / instruction):
    //   u   = bdtn * I
    //   inf = u * S                 (= beta*S*I/N * DT)
    //   sE  = sdt * E
    //   S  <- S - inf               (pk_fma with neg src0)
    //   E  <- E*(1-sdt) + inf
    //   I  <- I*(1-gdt) + sE
    // Staged across the 4 pairs so 4 independent ops issue back-to-back.
#pragma unroll 2
    for (int t = 0; t < SEIR_STEPS; ++t) {
        v2f u[NPAIR], inf[NPAIR], sE[NPAIR];
#pragma unroll
        for (int k = 0; k < NPAIR; ++k) u[k]   = pk_mul(bdtn[k], I[k]);
#pragma unroll
        for (int k = 0; k < NPAIR; ++k) inf[k] = pk_mul(u[k], S[k]);
#pragma unroll
        for (int k = 0; k < NPAIR; ++k) sE[k]  = pk_mul(sdt[k], E[k]);
#pragma unroll
        for (int k = 0; k < NPAIR; ++k) S[k]   = pk_fma_n0(u[k], S[k], S[k]);
#pragma unroll
        for (int k = 0; k < NPAIR; ++k) E[k]   = pk_fma(E[k], oms[k], inf[k]);
#pragma unroll
        for (int k = 0; k < NPAIR; ++k) I[k]   = pk_fma(I[k], omg[k], sE[k]);
    }

    // Only final infected count I is required by the reference.
    const float4 o_lo = make_float4(I[0].x, I[0].y, I[1].x, I[1].y);
    const float4 o_hi = make_float4(I[2].x, I[2].y, I[3].x, I[3].y);
    ((float4*)out)[base4]     = o_lo;
    ((float4*)out)[base4 + 1] = o_hi;
}

extern "C" void kernel_launch(void* const* d_in, const int* in_sizes, int n_in,
                              void* d_out, int out_size, void* d_ws, size_t ws_size,
                              hipStream_t stream) {
    // setup_inputs() order: beta, sigma, gamma, S0, E0, I0, R0 — all float32.
    const float* beta  = (const float*)d_in[0];
    const float* sigma = (const float*)d_in[1];
    const float* gamma = (const float*)d_in[2];
    const float* S0    = (const float*)d_in[3];
    const float* E0    = (const float*)d_in[4];
    const float* I0    = (const float*)d_in[5];
    // d_in[6] = R0: R never feeds back into S/E/I and the output is I only.

    float* out = (float*)d_out;
    const int B = in_sizes[0];           // 2,097,152 (divisible by 8)
    const int nthreads = B / 8;          // 8 trajectories per thread
    const int block = 256;
    const int grid = (nthreads + block - 1) / block;

    seir_pk_kernel<<<grid, block, 0, stream>>>(beta, sigma, gamma, S0, E0, I0,
                                               out, nthreads);
}